// SSIMLoss_833223655447
// MI455X (gfx1250) — compile-verified
//
#include <hip/hip_runtime.h>

typedef float v2f __attribute__((ext_vector_type(2)));
typedef float v8f __attribute__((ext_vector_type(8)));

#define IMG    512
#define OUTD   506           // 512 - 6 (VALID 7x7)
#define TILE   16
#define TIN    22            // TILE + 6
#define NTIL   32            // ceil(506/16)
#define BATCH  32
#define WPB    8             // waves per block (256 threads, wave32)
#define NWAVES (BATCH * NTIL * NTIL)   // 32768 wave-tiles
#define NBLK   (NWAVES / WPB)          // 4096

__global__ __launch_bounds__(256) void ssim_tile_kernel(
    const float* __restrict__ X, const float* __restrict__ Y,
    float* __restrict__ partial)
{
  const int lane   = threadIdx.x & 31;
  const int wv     = threadIdx.x >> 5;
  const int waveId = blockIdx.x * WPB + wv;
  const int b  = waveId >> 10;          // / (32*32)
  const int t  = waveId & 1023;
  const int r0 = (t >> 5) * TILE;
  const int c0 = (t & 31) * TILE;

  const float* __restrict__ xb = X + (size_t)b * IMG * IMG;
  const float* __restrict__ yb = Y + (size_t)b * IMG * IMG;

  // Per-wave LDS staging: 24 rows (22 + 2 zero K-pad) x 16 cols, fp32.
  // Used to transpose row-major horizontal sums into the WMMA B layout.
  __shared__ float4 hx4[WPB][24][4];
  float* hxf = (float*)&hx4[wv][0][0];

  // Zero the K-padding rows 22..23 once (never rewritten; DS in-order per wave).
  if (lane < 8) hx4[wv][22 + (lane >> 2)][lane & 3] = make_float4(0.f, 0.f, 0.f, 0.f);

  // ---- Load: lane l (<22) owns input row r0+l, columns c0..c0+21 ----
  const bool rowActive = lane < TIN;
  int rr = r0 + lane; if (rr > IMG - 1) rr = IMG - 1;   // edge tiles: masked later
  const float* px = xb + (size_t)rr * IMG;
  const float* py = yb + (size_t)rr * IMG;

  float xr[24], yr[24];
  if (rowActive) {
    if (c0 <= IMG - 24) {          // interior: 6x float4 per array (64B-aligned)
      #pragma unroll
      for (int i = 0; i < 6; ++i) {
        float4 vx4 = ((const float4*)(px + c0))[i];
        float4 vy4 = ((const float4*)(py + c0))[i];
        xr[4*i+0]=vx4.x; xr[4*i+1]=vx4.y; xr[4*i+2]=vx4.z; xr[4*i+3]=vx4.w;
        yr[4*i+0]=vy4.x; yr[4*i+1]=vy4.y; yr[4*i+2]=vy4.z; yr[4*i+3]=vy4.w;
      }
    } else {                       // last tile column: clamped scalar loads
      #pragma unroll
      for (int i = 0; i < TIN; ++i) {
        int cc = c0 + i; if (cc > IMG - 1) cc = IMG - 1;
        xr[i] = px[cc]; yr[i] = py[cc];
      }
    }
  }

  // ---- WMMA A operand: banded 0/1 vertical box matrix WV[m][k] = (m<=k<m+7) ----
  // fp32 16x16x4 A layout: lanes 0-15 / 16-31 both hold M=lane&15;
  // vgpr0 covers K = 4kc + 2*(lane>>4), vgpr1 covers K+1.
  const int m     = lane & 15;
  const int halfk = (lane >> 4) << 1;
  float A0[6], A1[6];
  #pragma unroll
  for (int kc = 0; kc < 6; ++kc) {
    int k0 = 4*kc + halfk;
    A0[kc] = (k0     >= m && k0     < m + 7) ? 1.f : 0.f;
    A1[kc] = (k0 + 1 >= m && k0 + 1 < m + 7) ? 1.f : 0.f;
  }

  v8f acc[5];
  #pragma unroll
  for (int q = 0; q < 5; ++q) {
    if (rowActive) {
      float tq[TIN];
      #pragma unroll
      for (int i = 0; i < TIN; ++i)
        tq[i] = (q == 0) ? xr[i]
              : (q == 1) ? yr[i]
              : (q == 2) ? xr[i] * xr[i]
              : (q == 3) ? yr[i] * yr[i]
              :            xr[i] * yr[i];
      // horizontal 7-tap box sum, sliding window
      float h[16];
      h[0] = ((tq[0] + tq[1]) + (tq[2] + tq[3])) + ((tq[4] + tq[5]) + tq[6]);
      #pragma unroll
      for (int c = 1; c < 16; ++c) h[c] = h[c-1] + tq[c+6] - tq[c-1];
      hx4[wv][lane][0] = make_float4(h[0],  h[1],  h[2],  h[3]);
      hx4[wv][lane][1] = make_float4(h[4],  h[5],  h[6],  h[7]);
      hx4[wv][lane][2] = make_float4(h[8],  h[9],  h[10], h[11]);
      hx4[wv][lane][3] = make_float4(h[12], h[13], h[14], h[15]);
    }
    // All 32 lanes fetch the B pattern (DS ops in-order within the wave).
    float b0[6], b1[6];
    #pragma unroll
    for (int kc = 0; kc < 6; ++kc) {
      int k0 = 4*kc + halfk;
      b0[kc] = hxf[ k0      * 16 + m];
      b1[kc] = hxf[(k0 + 1) * 16 + m];
    }
    // Vertical box sum: 6 chained fp32 WMMAs (K = 24, rows 22/23 are zero).
    v8f c = {};
    #pragma unroll
    for (int kc = 0; kc < 6; ++kc) {
      v2f a  = { A0[kc], A1[kc] };
      v2f bb = { b0[kc], b1[kc] };
      c = __builtin_amdgcn_wmma_f32_16x16x4_f32(false, a, false, bb,
                                                (short)0, c, false, false);
    }
    acc[q] = c;
  }

  // ---- SSIM map on the C/D layout: vgpr j -> M = j + 8*(lane>>4), N = lane&15 ----
  const float C1    = 1e-4f;          // (0.01*1.0)^2
  const float C2    = 9e-4f;          // (0.03*1.0)^2
  const float inv49 = 1.0f / 49.0f;
  const float covn  = 49.0f / 48.0f;  // N/(N-1)
  float lsum = 0.f;
  const int colg  = c0 + m;
  const int rbase = r0 + ((lane >> 4) << 3);
  #pragma unroll
  for (int j = 0; j < 8; ++j) {
    float ux  = acc[0][j] * inv49;
    float uy  = acc[1][j] * inv49;
    float uxx = acc[2][j] * inv49;
    float uyy = acc[3][j] * inv49;
    float uxy = acc[4][j] * inv49;
    float vx  = covn * (uxx - ux * ux);
    float vy  = covn * (uyy - uy * uy);
    float vxy = covn * (uxy - ux * uy);
    float num = (2.f * ux * uy + C1) * (2.f * vxy + C2);
    float den = (ux * ux + uy * uy + C1) * (vx + vy + C2);
    float S   = num / den;
    if ((rbase + j) < OUTD && colg < OUTD) lsum += S;
  }

  // wave32 reduction, then one partial per wave-tile (deterministic)
  #pragma unroll
  for (int off = 16; off > 0; off >>= 1)
    lsum += __shfl_xor(lsum, off, 32);
  if (lane == 0) partial[waveId] = lsum;
}

__global__ __launch_bounds__(256) void ssim_reduce_kernel(
    const float* __restrict__ partial, float* __restrict__ out, int n)
{
  __shared__ float sm[256];
  float s = 0.f;
  for (int i = threadIdx.x; i < n; i += 256) s += partial[i];  // fixed order
  sm[threadIdx.x] = s;
  __syncthreads();
  for (int off = 128; off > 0; off >>= 1) {
    if ((int)threadIdx.x < off) sm[threadIdx.x] += sm[threadIdx.x + off];
    __syncthreads();
  }
  if (threadIdx.x == 0)
    out[0] = 1.0f - sm[0] / (32.0f * 506.0f * 506.0f);
}

extern "C" void kernel_launch(void* const* d_in, const int* in_sizes, int n_in,
                              void* d_out, int out_size, void* d_ws, size_t ws_size,
                              hipStream_t stream) {
  const float* x = (const float*)d_in[0];
  const float* y = (const float*)d_in[1];
  float* partial = (float*)d_ws;        // NWAVES floats = 128 KB
  float* out     = (float*)d_out;       // single fp32 scalar

  ssim_tile_kernel<<<dim3(NBLK), dim3(256), 0, stream>>>(x, y, partial);
  ssim_reduce_kernel<<<dim3(1), dim3(256), 0, stream>>>(partial, out, NWAVES);
}